// Readout_68822555951732
// MI455X (gfx1250) — compile-verified
//
#include <hip/hip_runtime.h>

typedef float v2f __attribute__((ext_vector_type(2)));
typedef float v8f __attribute__((ext_vector_type(8)));
typedef unsigned int u32x4 __attribute__((ext_vector_type(4)));
typedef int          i32x4 __attribute__((ext_vector_type(4)));
typedef int          i32x8 __attribute__((ext_vector_type(8)));

#define D_FEAT 300
#define D4     75            // float4 per feature row in global memory
#define ROWF   304           // LDS row stride in floats (TDM zero-pads 300..303)
#define KC     24            // atoms staged per K-chunk (6 rows per wave TDM tile)
#define RPW    (KC / 4)      // rows per wave's TDM descriptor = 6
#define GPB    16            // graphs per block (WMMA M)
#define NTILE  ((D_FEAT + 15) / 16)   // 19 column tiles
#define NWAVE  4
#define TPW    ((NTILE + NWAVE - 1) / NWAVE)  // 5 tiles per wave max
#define THREADS (NWAVE * 32)

#if __has_builtin(__builtin_amdgcn_wmma_f32_16x16x4_f32)
#define HAVE_WMMA_F32X4 1
#else
#define HAVE_WMMA_F32X4 0
#endif

#if __has_builtin(__builtin_amdgcn_tensor_load_to_lds) && \
    __has_builtin(__builtin_amdgcn_s_wait_tensorcnt)
#define HAVE_TDM 1
#else
#define HAVE_TDM 0
#endif

#if HAVE_WMMA_F32X4

#if HAVE_TDM
// One 2-D TDM descriptor: copy up to RPW rows x 300 f32 (row stride 300) from
// global into LDS at lds_byte, writing 304 floats per LDS row (cols 300..303
// and rows past remRows are hardware zero-filled via tensor_dim OOB rules).
__device__ __forceinline__ void tdm_load_tile(const float* gsrc,
                                              unsigned lds_byte, int remRows) {
    unsigned long long ga = (unsigned long long)(uintptr_t)gsrc;
    int R = remRows < 0 ? 0 : remRows;

    u32x4 g0;
    g0[0] = 1u;                                            // count=1 (valid)
    g0[1] = lds_byte;                                      // lds_addr
    g0[2] = (unsigned)ga;                                  // global_addr[31:0]
    g0[3] = (unsigned)((ga >> 32) & 0x01FFFFFFull)         // global_addr[56:32]
          | 0x80000000u;                                   // type=2 ("image")

    i32x8 g1;
    g1[0] = 0x20000;                       // workgroup_mask=0, data_size=2 (4B)
    g1[1] = (D_FEAT & 0xFFFF) << 16;       // tensor_dim0[15:0] @ bits 63:48
    g1[2] = (R & 0xFFFF) << 16;            // tensor_dim0[31:16]=0 | tensor_dim1[15:0]
    g1[3] = ((R >> 16) & 0xFFFF)           // tensor_dim1[31:16]
          | (ROWF << 16);                  // tile_dim0 = 304
    g1[4] = RPW;                           // tile_dim1 = 6 (tile_dim2 = 0)
    g1[5] = D_FEAT;                        // tensor_dim0_stride[31:0] = 300
    g1[6] = 0;                             // stride0 hi | stride1 lo
    g1[7] = 0;

    i32x4 gz = (i32x4){0, 0, 0, 0};
#if __clang_major__ >= 23
    i32x8 gz8 = (i32x8){0, 0, 0, 0, 0, 0, 0, 0};
    __builtin_amdgcn_tensor_load_to_lds(g0, g1, gz, gz, gz8, 0);
#else
    __builtin_amdgcn_tensor_load_to_lds(g0, g1, gz, gz, 0);
#endif
}
#endif // HAVE_TDM

__global__ __launch_bounds__(THREADS)
void readout_wmma(const float* __restrict__ x,
                  const long long* __restrict__ scope,
                  float* __restrict__ out, int nG, int nA)
{
    __shared__ float sBuf[2][KC][ROWF];    // double-buffered staged atom rows
    __shared__ float sInv[GPB];
    __shared__ int   sLo[GPB], sHi[GPB];
    __shared__ int   sA[2];

    const int tid   = threadIdx.x;
    const int gbase = blockIdx.x * GPB;

    if (tid < GPB) {
        int g = gbase + tid;
        long long st = 0, sz = 0;
        if (g < nG) { st = scope[2 * g]; sz = scope[2 * g + 1]; }
        sLo[tid]  = (int)st;
        sHi[tid]  = (int)(st + sz);
        sInv[tid] = (sz > 0) ? 1.0f / (float)sz : 0.0f;
    }
    __syncthreads();
    if (tid == 0) {
        int lo = 0x7fffffff, hi = 0;
        for (int i = 0; i < GPB; ++i) {
            if (sHi[i] > sLo[i]) {
                lo = lo < sLo[i] ? lo : sLo[i];
                hi = hi > sHi[i] ? hi : sHi[i];
            }
        }
        if (hi <= lo) { lo = 0; hi = 0; }
        sA[0] = lo; sA[1] = hi;
    }
    __syncthreads();

    const int aLo = sA[0], aHi = sA[1];
    const int nChunk = (aHi - aLo + KC - 1) / KC;

    const int wv   = __builtin_amdgcn_readfirstlane((int)(tid >> 5)); // scalar wave id
    const int lane = tid & 31;
    const int m    = lane & 15;     // A-matrix M row / B/N column within tile
    const int hiH  = lane >> 4;     // half-wave: K pair {0,1} vs {2,3}

    v8f acc[TPW];
    for (int t = 0; t < TPW; ++t) acc[t] = (v8f){};

#if HAVE_TDM
    // Preload chunk 0: each wave DMAs its RPW-row slice into buffer 0.
    if (nChunk > 0) {
        int rowStart = aLo + wv * RPW;
        unsigned lb = (unsigned)(uintptr_t)&sBuf[0][wv * RPW][0];
        tdm_load_tile(x + (long long)rowStart * D_FEAT, lb, nA - rowStart);
    }
#endif

    for (int ch = 0; ch < nChunk; ++ch) {
        const int rowBase = aLo + ch * KC;
        const int b = ch & 1;

#if HAVE_TDM
        __builtin_amdgcn_s_wait_tensorcnt(0);   // my slice of chunk ch landed
        __syncthreads();                        // all waves' slices landed
        if (ch + 1 < nChunk) {                  // overlap: DMA chunk ch+1
            int rowStart = aLo + (ch + 1) * KC + wv * RPW;
            unsigned lb = (unsigned)(uintptr_t)&sBuf[(ch + 1) & 1][wv * RPW][0];
            tdm_load_tile(x + (long long)rowStart * D_FEAT, lb, nA - rowStart);
        }
#else
        // Fallback staging: cooperative float4 loads with zero fill.
        for (int i = tid; i < KC * (ROWF / 4); i += THREADS) {
            int r = i / (ROWF / 4);
            int c = i - r * (ROWF / 4);
            int row = rowBase + r;
            float4 v = make_float4(0.f, 0.f, 0.f, 0.f);
            if (c < D4 && row < aHi && row < nA)
                v = ((const float4*)x)[(long long)row * D4 + c];
            ((float4*)&sBuf[b][r][0])[c] = v;
        }
        __syncthreads();
#endif

        // ---- A operand: membership weights (16x4 f32 A layout) ----
        v2f wv4[KC / 4];
        const int lo = sLo[m], hi = sHi[m];
        for (int k4 = 0; k4 < KC / 4; ++k4) {
            int k0 = rowBase + k4 * 4 + (hiH ? 2 : 0);
            wv4[k4].x = (k0     >= lo && k0     < hi) ? 1.0f : 0.0f;
            wv4[k4].y = (k0 + 1 >= lo && k0 + 1 < hi) ? 1.0f : 0.0f;
        }

        // ---- WMMA accumulate over this chunk for each owned column tile ----
        for (int t = 0; t < TPW; ++t) {
            int tile = wv + NWAVE * t;
            if (tile >= NTILE) break;
            int nb = tile * 16;
            v8f a = acc[t];
            for (int k4 = 0; k4 < KC / 4; ++k4) {
                int kr = k4 * 4 + (hiH ? 2 : 0);
                v2f bb;
                bb.x = sBuf[b][kr][nb + m];
                bb.y = sBuf[b][kr + 1][nb + m];
                a = __builtin_amdgcn_wmma_f32_16x16x4_f32(
                        false, wv4[k4], false, bb, (short)0, a, false, false);
            }
            acc[t] = a;
        }
        __syncthreads();   // compute done before this buffer is overwritten
    }

    // ---- epilogue: scale by 1/size, store (C/D: VGPR e -> M=e or e+8) ----
    for (int t = 0; t < TPW; ++t) {
        int tile = wv + NWAVE * t;
        if (tile >= NTILE) break;
        int col = tile * 16 + m;
        if (col >= D_FEAT) continue;
        for (int e = 0; e < 8; ++e) {
            int M = e + (hiH ? 8 : 0);
            int g = gbase + M;
            if (g < nG)
                out[(long long)g * D_FEAT + col] = acc[t][e] * sInv[M];
        }
    }
}
#endif // HAVE_WMMA_F32X4

// Fallback (only if f32 WMMA builtin absent): plain streaming mean.
__global__ __launch_bounds__(128)
void readout_valu(const float* __restrict__ x,
                  const long long* __restrict__ scope,
                  float* __restrict__ out, int nG, int nA)
{
    int g = blockIdx.x;
    if (g >= nG) return;
    long long st = scope[2 * g];
    long long sz = scope[2 * g + 1];
    float inv = (sz > 0) ? 1.0f / (float)sz : 0.0f;
    for (int c = threadIdx.x; c < D_FEAT; c += blockDim.x) {
        float s = 0.0f;
        long long e = st + sz;
        for (long long r = st; r < e; ++r)
            if (r < nA) s += x[r * D_FEAT + c];
        out[(long long)g * D_FEAT + c] = s * inv;
    }
}

extern "C" void kernel_launch(void* const* d_in, const int* in_sizes, int n_in,
                              void* d_out, int out_size, void* d_ws, size_t ws_size,
                              hipStream_t stream)
{
    const float*     x     = (const float*)d_in[0];
    const long long* scope = (const long long*)d_in[1];   // int64 (start,size) pairs
    float*           out   = (float*)d_out;

    int nA = in_sizes[0] / D_FEAT;
    int nG = in_sizes[1] / 2;

#if HAVE_WMMA_F32X4
    int blocks = (nG + GPB - 1) / GPB;
    hipLaunchKernelGGL(readout_wmma, dim3(blocks), dim3(THREADS), 0, stream,
                       x, scope, out, nG, nA);
#else
    hipLaunchKernelGGL(readout_valu, dim3(nG), dim3(128), 0, stream,
                       x, scope, out, nG, nA);
#endif
}